// GCN_44049184588268
// MI455X (gfx1250) — compile-verified
//
#include <hip/hip_runtime.h>
#include <hip/hip_bf16.h>
#include <math.h>

typedef float v2f __attribute__((ext_vector_type(2)));
typedef float v8f __attribute__((ext_vector_type(8)));

#define GCN_NHID 256
#define GCN_EPW  256
#define GEMM_KC  32          // K-slab staged in LDS per iteration
#define WPAD     264         // 256 + 8: rows shift 16 banks -> conflict-free frags

// ---------------------------------------------------------------------------
// fp32 WMMA GEMM: C[M,N] = A[M,K] * W[K,N],  N == 256.
// Block = 256 threads = 8 waves = 4 row-strips x 2 half-strips.
// Each wave owns a 16x128 output strip: 8 N-tiles, 8x v8f accumulators.
// Per k-step: one A-frag load (b64, reused 8x), B-frags from an LDS-staged
// W slab loaded with CDNA5 async-to-LDS (ASYNCcnt + s_wait_asynccnt).
// Frag layouts per ISA 7.12.2:
//   A(16x4): m=lane&15, k = k0 + 2*(lane>>4) + {0,1}
//   B(4x16): n=lane&15, k = k0 + 2*(lane>>4) + {0,1}
//   C(16x16): VGPR v -> m = v + 8*(lane>>4), n = lane&15
// ---------------------------------------------------------------------------
__global__ __launch_bounds__(256) void gcn_gemm_f32_wmma(
    const float* __restrict__ A, const float* __restrict__ W,
    float* __restrict__ C, int M, int K) {
  __shared__ float wslab[GEMM_KC * WPAD];   // 33.8 KB of the 320 KB WGP LDS

  const int lane   = threadIdx.x & 31;
  const int waveIB = threadIdx.x >> 5;            // 0..7
  const int half   = waveIB & 1;                  // which 128-col half
  const int tileM  = blockIdx.x * 4 + (waveIB >> 1);
  const int tilesM = M >> 4;                      // M % 16 == 0 here
  const bool active = (tileM < tilesM);           // wave-uniform

  const int kh   = (lane >> 4) << 1;              // 0 or 2
  const int nloc = (lane & 15);
  const int mrow = tileM * 16 + nloc;

  v8f acc[8];
#pragma unroll
  for (int t = 0; t < 8; ++t) acc[t] = (v8f){0.f,0.f,0.f,0.f,0.f,0.f,0.f,0.f};

  const float* __restrict__ Ap = A + (size_t)mrow * K + kh;

  for (int kc = 0; kc < K; kc += GEMM_KC) {
    // ---- cooperative async W-slab load: 32 rows x 256 cols, b128 chunks ----
    // 2048 16B-chunks / 256 threads = 8 chunks each.
#pragma unroll
    for (int i = 0; i < 8; ++i) {
      const int chunk = threadIdx.x + (i << 8);
      const int r = chunk >> 6;                   // 0..31
      const int q = chunk & 63;                   // 0..63 quad-col
      const float* gsrc = W + (size_t)(kc + r) * GCN_NHID + (q << 2);
      const unsigned ldsAddr =
          (unsigned)(size_t)(&wslab[r * WPAD + (q << 2)]);
      asm volatile("global_load_async_to_lds_b128 %0, %1, off"
                   :: "v"(ldsAddr), "v"(gsrc) : "memory");
    }
    asm volatile("s_wait_asynccnt 0x0" ::: "memory");
    __syncthreads();

    if (active) {
#pragma unroll 2
      for (int kk = 0; kk < GEMM_KC; kk += 4) {
        v2f a;
        a.x = Ap[kc + kk];                        // contiguous pair -> b64
        a.y = Ap[kc + kk + 1];
        const float* __restrict__ ws0 = &wslab[(kk + kh) * WPAD];
        const float* __restrict__ ws1 = ws0 + WPAD;
#pragma unroll
        for (int t = 0; t < 8; ++t) {
          const int ncol = ((half << 3) + t) * 16 + nloc;
          v2f b;
          b.x = ws0[ncol];                        // ds_load_b32, conflict-free
          b.y = ws1[ncol];
          acc[t] = __builtin_amdgcn_wmma_f32_16x16x4_f32(
              false, a, false, b, (short)0, acc[t], false, false);
        }
      }
    }
    __syncthreads();                              // slab reuse fence
  }

  if (active) {
    const int rbase = tileM * 16 + ((lane >> 4) << 3);
#pragma unroll
    for (int t = 0; t < 8; ++t) {
      const int ncol = ((half << 3) + t) * 16 + nloc;
#pragma unroll
      for (int v = 0; v < 8; ++v)
        C[(size_t)(rbase + v) * GCN_NHID + ncol] = acc[t][v];
    }
  }
}

// ---------------------------------------------------------------------------
// Zero-fill (scatter targets re-zeroed every call; ws is not re-poisoned)
// ---------------------------------------------------------------------------
__global__ void gcn_zero_f32(float4* __restrict__ p, long n4) {
  long i = (long)blockIdx.x * blockDim.x + threadIdx.x;
  long stride = (long)gridDim.x * blockDim.x;
  float4 z = {0.f, 0.f, 0.f, 0.f};
  for (; i < n4; i += stride) p[i] = z;
}

// ---------------------------------------------------------------------------
// SpMM (COO, rows sorted): out[r,:] += w_e * h[c_e,:]
// One wave per 256 consecutive edges over ALL 256 features (8 acc/lane).
// Edge triples loaded coalesced in 32-batches, broadcast via __shfl;
// register segment accumulation, atomic flush at segment boundaries only
// (avg segment = E/N = 32 edges). Gathers are 1KB contiguous, L2-resident.
// ---------------------------------------------------------------------------
__global__ void gcn_spmm(const float* __restrict__ h,
                         const int*   __restrict__ erow,
                         const int*   __restrict__ ecol,
                         const float* __restrict__ ew,
                         float* __restrict__ out,
                         long E) {
  const int lane = threadIdx.x & 31;
  const long wave = ((long)blockIdx.x * blockDim.x + threadIdx.x) >> 5;
  const long e0 = wave * GCN_EPW;
  if (e0 >= E) return;
  const long e1 = (e0 + GCN_EPW < E) ? (e0 + GCN_EPW) : E;

  float acc[8];
#pragma unroll
  for (int v = 0; v < 8; ++v) acc[v] = 0.f;
  int curRow = -1;

  for (long base = e0; base < e1; base += 32) {
    const long idx = base + lane;
    int r = 0, cI = 0; float wv = 0.f;
    if (idx < e1) {
      r  = erow[idx];
      cI = ecol[idx];
      wv = ew[idx];
      __builtin_prefetch(&h[(size_t)cI * GCN_NHID], 0, 0);  // global_prefetch_b8
    }
    const int cnt = (int)(((e1 - base) < 32) ? (e1 - base) : 32);
    for (int j = 0; j < cnt; ++j) {
      const int   rj = __shfl(r,  j);
      const int   cj = __shfl(cI, j);
      const float wj = __shfl(wv, j);
      if (rj != curRow) {
        if (curRow >= 0) {
          float* op = out + (size_t)curRow * GCN_NHID + lane;
#pragma unroll
          for (int v = 0; v < 8; ++v) { atomicAdd(op + v * 32, acc[v]); acc[v] = 0.f; }
        }
        curRow = rj;
      }
      const float* __restrict__ hp = h + (size_t)cj * GCN_NHID + lane;
#pragma unroll
      for (int v = 0; v < 8; ++v) acc[v] = fmaf(wj, hp[v * 32], acc[v]);
    }
  }
  if (curRow >= 0) {
    float* op = out + (size_t)curRow * GCN_NHID + lane;
#pragma unroll
    for (int v = 0; v < 8; ++v) atomicAdd(op + v * 32, acc[v]);
  }
}

// ---------------------------------------------------------------------------
// o = relu(in + b[c]) * (u < 0.5 ? 2 : 0)    (bias + relu + dropout, fused)
// ---------------------------------------------------------------------------
__global__ void gcn_bias_relu_drop(const float* __restrict__ in,
                                   const float* __restrict__ b,
                                   const float* __restrict__ u,
                                   float* __restrict__ o,
                                   long total) {
  long i = (long)blockIdx.x * blockDim.x + threadIdx.x;
  long stride = (long)gridDim.x * blockDim.x;
  for (; i < total; i += stride) {
    const int c = (int)(i & (GCN_NHID - 1));
    float v = in[i] + b[c];
    v = v > 0.f ? v : 0.f;
    o[i] = (u[i] < 0.5f) ? v * 2.0f : 0.f;
  }
}

// ---------------------------------------------------------------------------
// out = log_softmax(relu(in + b2), axis=1). One wave per 256-wide row,
// 8 elems/lane; max & sum reductions via 5-step __shfl_xor (wave32).
// ---------------------------------------------------------------------------
__global__ void gcn_bias_relu_logsoftmax(const float* __restrict__ in,
                                         const float* __restrict__ b2,
                                         float* __restrict__ out,
                                         int M) {
  const int lane = threadIdx.x & 31;
  const int row  = (int)((blockIdx.x * blockDim.x + threadIdx.x) >> 5);
  if (row >= M) return;

  const float* __restrict__ ip = in + (size_t)row * GCN_NHID + lane;
  float t[8];
  float m = -1e30f;
#pragma unroll
  for (int v = 0; v < 8; ++v) {
    float x = ip[v * 32] + b2[v * 32 + lane];
    x = x > 0.f ? x : 0.f;
    t[v] = x;
    m = fmaxf(m, x);
  }
#pragma unroll
  for (int off = 16; off > 0; off >>= 1) m = fmaxf(m, __shfl_xor(m, off));

  float s = 0.f;
#pragma unroll
  for (int v = 0; v < 8; ++v) s += __expf(t[v] - m);
#pragma unroll
  for (int off = 16; off > 0; off >>= 1) s += __shfl_xor(s, off);
  const float lse = __logf(s) + m;

  float* __restrict__ op = out + (size_t)row * GCN_NHID + lane;
#pragma unroll
  for (int v = 0; v < 8; ++v) op[v * 32] = t[v] - lse;
}

// ---------------------------------------------------------------------------
extern "C" void kernel_launch(void* const* d_in, const int* in_sizes, int n_in,
                              void* d_out, int out_size, void* d_ws, size_t ws_size,
                              hipStream_t stream) {
  const float* x    = (const float*)d_in[0];
  const float* W1   = (const float*)d_in[1];
  const float* b1   = (const float*)d_in[2];
  const float* W2   = (const float*)d_in[3];
  const float* b2   = (const float*)d_in[4];
  const int*   erow = (const int*)d_in[5];
  const int*   ecol = (const int*)d_in[6];
  const float* ew   = (const float*)d_in[7];
  const float* du   = (const float*)d_in[8];
  float* outp = (float*)d_out;

  const int NHID  = in_sizes[2];            // 256
  const int NFEAT = in_sizes[1] / NHID;     // 512
  const int N     = in_sizes[0] / NFEAT;    // 100000
  const long E    = in_sizes[5];            // 3,200,000
  const long HN   = (long)N * NHID;         // 25.6M elements

  float* bufA = (float*)d_ws;               // 102.4 MB
  float* bufB = bufA + HN;                  // 102.4 MB

  const int BT = 256;                       // 8 waves / block
  const int tilesM     = N / 16;            // 6250
  const int gemmBlocks = (tilesM + 3) / 4;  // 4 row-strips per block
  const int zeroBlocks = 2048;
  const long spmmWaves = (E + GCN_EPW - 1) / GCN_EPW;
  const int  spmmBlocks = (int)((spmmWaves * 32 + BT - 1) / BT);
  const int  elemBlocks = 4096;
  const int  lsBlocks   = (int)(((long)N * 32 + BT - 1) / BT);

  // 1) A = x @ W1                      (WMMA fp32, async-LDS W staging)
  gcn_gemm_f32_wmma<<<gemmBlocks, BT, 0, stream>>>(x, W1, bufA, N, NFEAT);
  // 2) B = 0 ; 3) B += scatter(A)      (SpMM layer 1)
  gcn_zero_f32<<<zeroBlocks, BT, 0, stream>>>((float4*)bufB, HN / 4);
  gcn_spmm<<<spmmBlocks, BT, 0, stream>>>(bufA, erow, ecol, ew, bufB, E);
  // 4) A = relu(B + b1) * drop
  gcn_bias_relu_drop<<<elemBlocks, BT, 0, stream>>>(bufB, b1, du, bufA, HN);
  // 5) B = A @ W2                      (WMMA fp32)
  gcn_gemm_f32_wmma<<<gemmBlocks, BT, 0, stream>>>(bufA, W2, bufB, N, NHID);
  // 6) A = 0 ; 7) A += scatter(B)      (SpMM layer 2)
  gcn_zero_f32<<<zeroBlocks, BT, 0, stream>>>((float4*)bufA, HN / 4);
  gcn_spmm<<<spmmBlocks, BT, 0, stream>>>(bufB, erow, ecol, ew, bufA, E);
  // 8) out = log_softmax(relu(A + b2))
  gcn_bias_relu_logsoftmax<<<lsBlocks, BT, 0, stream>>>(bufA, b2, outp, N);
}